// Token_Selective_Attention_12214886990428
// MI455X (gfx1250) — compile-verified
//
#include <hip/hip_runtime.h>
#include <hip/hip_bf16.h>
#include <math.h>

typedef __attribute__((ext_vector_type(2))) float    v2f;
typedef __attribute__((ext_vector_type(4))) float    v4f;
typedef __attribute__((ext_vector_type(8))) float    v8f;
typedef __attribute__((ext_vector_type(4))) unsigned v4u;

#define NTOK   4096      // h*w*t tokens per head
#define KTOP   3276      // int(4096*0.8)
#define NHEADS 8
// per-head layouts: arr[head][token][4 channels] (16B per token, fully vectorizable)

__device__ __forceinline__ unsigned f2key(float x) {
    unsigned u = __float_as_uint(x);
    return (u & 0x80000000u) ? ~u : (u | 0x80000000u);   // monotone float key
}
__device__ __forceinline__ float wred_sum(float v) {
    #pragma unroll
    for (int off = 16; off; off >>= 1) v += __shfl_xor(v, off, 32);
    return v;
}
__device__ __forceinline__ float wred_max(float v) {
    #pragma unroll
    for (int off = 16; off; off >>= 1) v = fmaxf(v, __shfl_xor(v, off, 32));
    return v;
}

// ---------- K1: 1x1x1 group conv: lin[o][c][s] = sum_t wq[o*4+t] * x[(t*32+c)*1024+s]
__global__ __launch_bounds__(256)
void qkv_lin_kernel(const float* __restrict__ x, const float* __restrict__ wq,
                    float* __restrict__ lin) {
    int idx = blockIdx.x * 256 + threadIdx.x;      // 12*32*1024
    if (idx >= 12 * 32 * 1024) return;
    int s = idx & 1023;
    int c = (idx >> 10) & 31;
    int o = idx >> 15;
    float acc = 0.f;
    #pragma unroll
    for (int t = 0; t < 4; t++)
        acc = fmaf(wq[o * 4 + t], x[(t * 32 + c) * 1024 + s], acc);
    lin[idx] = acc;
}

// ---------- K2: depthwise 3x3 (weights shared over C); 4 channels (one head) per thread;
// scatter into per-head token-major layout: dst[hd][(y*32+x)*4+tt][cc]  (one v4f store)
__global__ __launch_bounds__(256)
void qkv_dw_kernel(const float* __restrict__ lin, const float* __restrict__ wdw,
                   float* __restrict__ qh, float* __restrict__ kh, float* __restrict__ vh) {
    int idx = blockIdx.x * 256 + threadIdx.x;      // 12*8*1024
    if (idx >= 12 * 8 * 1024) return;
    int s  = idx & 1023;
    int hd = (idx >> 10) & 7;
    int o  = idx >> 13;
    int yy = s >> 5, xx = s & 31;
    float w[9];
    #pragma unroll
    for (int i = 0; i < 9; i++) w[i] = wdw[o * 9 + i];
    v4f acc = {0.f, 0.f, 0.f, 0.f};
    #pragma unroll
    for (int dy = -1; dy <= 1; dy++) {
        int y2 = yy + dy;
        if ((unsigned)y2 >= 32u) continue;
        #pragma unroll
        for (int dx = -1; dx <= 1; dx++) {
            int x2 = xx + dx;
            if ((unsigned)x2 >= 32u) continue;
            float wv = w[(dy + 1) * 3 + (dx + 1)];
            int pix = y2 * 32 + x2;
            #pragma unroll
            for (int cc = 0; cc < 4; cc++)
                acc[cc] = fmaf(wv, lin[(o * 32 + hd * 4 + cc) * 1024 + pix], acc[cc]);
        }
    }
    int tt = o & 3;
    float* dst = (o < 4) ? qh : ((o < 8) ? kh : vh);
    *(v4f*)(dst + hd * (NTOK * 4) + (s * 4 + tt) * 4) = acc;
}

// ---------- K3: per-channel sum of squares (coalesced; 2 blocks: q then k)
__global__ __launch_bounds__(1024)
void sumsq_kernel(const float* __restrict__ qh, const float* __restrict__ kh,
                  float* __restrict__ norms) {
    __shared__ float red[1024];
    const float* src = blockIdx.x ? kh : qh;
    int t  = threadIdx.x;
    int hd = t >> 7;           // 0..7
    int l  = t & 127;
    int cc = l & 3;
    int g  = l >> 2;           // token group 0..31
    const float* base = src + hd * (NTOK * 4);
    float s = 0.f;
    for (int k = 0; k < NTOK / 32; k++) {
        float v = base[(g + 32 * k) * 4 + cc];
        s = fmaf(v, v, s);
    }
    red[t] = s;
    __syncthreads();
    for (int st = 64; st >= 4; st >>= 1) {
        if ((t & 127) < st) red[t] += red[t + st];
        __syncthreads();
    }
    if ((t & 127) < 4)
        norms[blockIdx.x * 32 + hd * 4 + cc] = red[t];
}

// ---------- K4: fused attention tile: WMMA logits -> radix top-k -> softmax -> xV
// grid (256 row-tiles, 8 heads), 256 thr, dyn LDS: 16x4096 f32 logits + hists + sel
__global__ __launch_bounds__(256)
void attn_kernel(const float* __restrict__ qh, const float* __restrict__ kh,
                 const float* __restrict__ vh, const float* __restrict__ norms,
                 const float* __restrict__ temp, float* __restrict__ oh) {
    extern __shared__ float lds[];
    float*    logits = lds;                               // 16*4096 f32 (256 KB)
    unsigned* hist   = (unsigned*)(lds + 16 * NTOK);      // 8 waves * 256 bins
    unsigned* sel    = hist + 8 * 256;                    // 2 per wave

    const int head = blockIdx.y;
    const int n0   = blockIdx.x * 16;
    const int wave = threadIdx.x >> 5;
    const int lane = threadIdx.x & 31;

    const float* qb = qh + head * (NTOK * 4);
    const float* kb = kh + head * (NTOK * 4);
    const float* vb = vh + head * (NTOK * 4);
    const float  temperature = temp[head];

    // temperature folded into Q scales (positive scalar: top-k order unchanged,
    // softmax sees identical logits as the reference)
    float qs[4], ks[4];
    #pragma unroll
    for (int c = 0; c < 4; c++) {
        qs[c] = temperature / fmaxf(sqrtf(norms[head * 4 + c]), 1e-12f);
        ks[c] = 1.0f / fmaxf(sqrtf(norms[32 + head * 4 + c]), 1e-12f);
    }

    // A fragment (Q tile 16x4): f32 WMMA layout: lanes<16 hold K0,K1; lanes>=16 K2,K3
    const int col = lane & 15;
    const int kb2 = (lane >> 4) * 2;
    v2f a = *(const v2f*)(qb + (n0 + col) * 4 + kb2);     // one b64 load
    a.x *= qs[kb2 + 0];
    a.y *= qs[kb2 + 1];
    const float ks0 = ks[kb2 + 0], ks1 = ks[kb2 + 1];
    const int rbase = (lane >> 4) * 8;

    // logits via V_WMMA_F32_16X16X4_F32, m-tiles round-robin over the 8 waves
    for (int mt = wave; mt < NTOK / 16; mt += 8) {
        const int m0 = mt * 16;
        __builtin_prefetch(kb + (m0 + 128) * 4, 0, 3);    // WGP-scope, next tile
        v2f bf = *(const v2f*)(kb + (m0 + col) * 4 + kb2);
        bf.x *= ks0;
        bf.y *= ks1;
        v8f cz = {};
        v8f d = __builtin_amdgcn_wmma_f32_16x16x4_f32(false, a, false, bf,
                                                      (short)0, cz, false, false);
        #pragma unroll
        for (int r = 0; r < 8; r++)
            logits[(r + rbase) * NTOK + m0 + col] = d[r];
    }
    __syncthreads();

    // per-row: exact kk-th-largest via 4x8-bit radix select, then softmax + xV
    unsigned* whist = hist + wave * 256;
    for (int r2 = 0; r2 < 2; r2++) {
        const int row = wave * 2 + r2;
        const float* L = logits + row * NTOK;

        float mx = -INFINITY;
        for (int mb = lane * 4; mb < NTOK; mb += 128) {   // b128 LDS reads
            v4f l4 = *(const v4f*)(L + mb);
            mx = fmaxf(mx, fmaxf(fmaxf(l4.x, l4.y), fmaxf(l4.z, l4.w)));
        }
        mx = wred_max(mx);

        unsigned prefix = 0, remaining = KTOP;
        for (int pass = 0; pass < 4; pass++) {
            v4u z = {0u, 0u, 0u, 0u};
            *(v4u*)(whist + lane * 8)     = z;            // zero 8 bins / lane
            *(v4u*)(whist + lane * 8 + 4) = z;
            __syncthreads();
            const int dsh = 24 - 8 * pass;
            for (int mb = lane * 4; mb < NTOK; mb += 128) {
                v4f l4 = *(const v4f*)(L + mb);
                #pragma unroll
                for (int j = 0; j < 4; j++) {
                    unsigned key = f2key(l4[j]);
                    bool match = (pass == 0) || ((key >> (32 - 8 * pass)) == prefix);
                    if (match) atomicAdd(&whist[(key >> dsh) & 255u], 1u);
                }
            }
            __syncthreads();
            // parallel digit resolve: lane owns bins [lane*8, lane*8+8)
            v4u h0 = *(const v4u*)(whist + lane * 8);     // 2x ds_load_b128
            v4u h1 = *(const v4u*)(whist + lane * 8 + 4);
            unsigned bl[8] = {h0.x, h0.y, h0.z, h0.w, h1.x, h1.y, h1.z, h1.w};
            unsigned lsum = 0;
            #pragma unroll
            for (int j = 0; j < 8; j++) lsum += bl[j];
            unsigned cum = lsum;                          // suffix-inclusive scan
            #pragma unroll
            for (int off = 1; off < 32; off <<= 1) {
                unsigned othr = __shfl_down(cum, off, 32);
                if (lane + off < 32) cum += othr;
            }
            unsigned cumAbove = cum - lsum;               // keys in strictly higher bins
            if (remaining > cumAbove && remaining <= cumAbove + lsum) {
                unsigned c = cumAbove;
                int digit = -1;
                #pragma unroll
                for (int j = 7; j >= 0; j--) {
                    if (digit < 0) {
                        if (c + bl[j] >= remaining) digit = lane * 8 + j;
                        else c += bl[j];
                    }
                }
                sel[wave * 2 + 0] = (prefix << 8) | (unsigned)digit;
                sel[wave * 2 + 1] = remaining - c;
            }
            __syncthreads();
            prefix    = sel[wave * 2 + 0];
            remaining = sel[wave * 2 + 1];
        }
        const unsigned tkey = prefix;   // key of kk-th largest; keep key >= tkey

        float sumE = 0.f, a0 = 0.f, a1 = 0.f, a2 = 0.f, a3 = 0.f;
        for (int mb = lane * 4; mb < NTOK; mb += 128) {
            v4f l4 = *(const v4f*)(L + mb);               // b128 LDS read
            #pragma unroll
            for (int j = 0; j < 4; j++) {
                float l = l4[j];
                if (f2key(l) >= tkey) {
                    float p = __expf(l - mx);
                    sumE += p;
                    v4f vv = *(const v4f*)(vb + (mb + j) * 4);  // one b128 V load
                    a0 = fmaf(p, vv.x, a0);
                    a1 = fmaf(p, vv.y, a1);
                    a2 = fmaf(p, vv.z, a2);
                    a3 = fmaf(p, vv.w, a3);
                }
            }
        }
        sumE = wred_sum(sumE);
        a0 = wred_sum(a0); a1 = wred_sum(a1); a2 = wred_sum(a2); a3 = wred_sum(a3);
        if (lane == 0) {
            float inv = 1.0f / sumE;
            v4f o4 = {a0 * inv, a1 * inv, a2 * inv, a3 * inv};
            *(v4f*)(oh + head * (NTOK * 4) + (n0 + row) * 4) = o4;  // one b128 store
        }
        __syncthreads();
    }
}

// ---------- K5: un-permute + 128x128 projection via WMMA f32 16x16x4 (K looped x32)
// out[o*1024+s] = sum_d wproj[o*128+d] * X[d][s]
// X[d][s] = oh[hd][(s*4+tt)*4 + cc], d = tt*32 + hd*4 + cc
__global__ __launch_bounds__(256)
void proj_kernel(const float* __restrict__ oh, const float* __restrict__ wproj,
                 float* __restrict__ out) {
    const int wave = threadIdx.x >> 5;
    const int lane = threadIdx.x & 31;
    const int tile = blockIdx.x * 8 + wave;    // 512 tiles: 8 (o) x 64 (s)
    const int o0 = (tile >> 6) * 16;
    const int s0 = (tile & 63) * 16;
    const int col = lane & 15;
    const int kb2 = (lane >> 4) * 2;

    v8f d = {};
    for (int kc = 0; kc < 32; kc++) {
        const int c0 = kc * 4 + kb2;                        // even -> cc in {0,2}
        v2f a = *(const v2f*)(wproj + (o0 + col) * 128 + c0);
        const int tt = c0 >> 5;
        const int hd = (c0 >> 2) & 7;
        const int cc = c0 & 3;
        v2f b = *(const v2f*)(oh + hd * (NTOK * 4) + ((s0 + col) * 4 + tt) * 4 + cc);
        d = __builtin_amdgcn_wmma_f32_16x16x4_f32(false, a, false, b,
                                                  (short)0, d, false, false);
    }
    const int rbase = (lane >> 4) * 8;
    #pragma unroll
    for (int r = 0; r < 8; r++)
        out[(o0 + r + rbase) * 1024 + s0 + col] = d[r];
}

extern "C" void kernel_launch(void* const* d_in, const int* in_sizes, int n_in,
                              void* d_out, int out_size, void* d_ws, size_t ws_size,
                              hipStream_t stream) {
    const float* x     = (const float*)d_in[0];   // [1,128,32,32]
    const float* temp  = (const float*)d_in[1];   // [1,8,1,1]
    const float* wqkv  = (const float*)d_in[2];   // [12,4,1,1,1]
    const float* wdw   = (const float*)d_in[3];   // [12,1,1,3,3]
    const float* wproj = (const float*)d_in[4];   // [128,128,1,1]

    float* ws    = (float*)d_ws;
    float* lin   = ws;                 // 12*32*1024 = 393216
    float* qh    = lin + 393216;       // 8*4096*4  = 131072 (per-head token-major)
    float* kh    = qh + 131072;
    float* vh    = kh + 131072;
    float* oh    = vh + 131072;        // attention output heads (same layout)
    float* norms = oh + 131072;        // 64 (32 q + 32 k)

    qkv_lin_kernel<<<1536, 256, 0, stream>>>(x, wqkv, lin);
    qkv_dw_kernel <<<384,  256, 0, stream>>>(lin, wdw, qh, kh, vh);
    sumsq_kernel  <<<2,   1024, 0, stream>>>(qh, kh, norms);

    size_t ldsbytes = (size_t)16 * NTOK * 4      // logits tile (256 KB)
                    + (size_t)8 * 256 * 4        // per-wave histograms
                    + (size_t)16 * 4;            // select scratch
    attn_kernel<<<dim3(NTOK / 16, NHEADS), 256, ldsbytes, stream>>>(
        qh, kh, vh, norms, temp, oh);

    proj_kernel<<<64, 256, 0, stream>>>(oh, wproj, (float*)d_out);
}